// sGravity_21792664059967
// MI455X (gfx1250) — compile-verified
//
#include <hip/hip_runtime.h>
#include <math.h>

typedef float v2f __attribute__((ext_vector_type(2)));
typedef float v8f __attribute__((ext_vector_type(8)));

#define Bn   64
#define Cc   64
#define Nn   325
#define Tt   12
#define NT   (Nn * Tt)     // 3900 columns per batch in pooling GEMM
#define ROWS (Bn * Tt * Nn) // 249600 softmax rows
#define TILES_PER_B ((NT + 15) / 16) // 244

__device__ __forceinline__ float softplus_f(float x) {
    // logaddexp(x, 0) = max(x,0) + log1p(exp(-|x|))
    return fmaxf(x, 0.0f) + log1pf(expf(-fabsf(x)));
}

// ---------------------------------------------------------------------------
// Phase 1: pooling as P[2, 3900] = W[2,64] @ X_b[64,3900] via WMMA f32 16x16x4,
// then softplus+1, pow, fold g. One wave per 16-column tile per batch b.
// A-matrix (16x4 f32): VGPR v, lane l -> (M = l%16, K = v + 2*(l/16)).
// B-matrix (4x16 f32): VGPR v, lane l -> (K = v + 2*(l/16), N = l%16).
// D (16x16 f32):       VGPR r, lane l -> (M = r + 8*(l/16), N = l%16).
// ---------------------------------------------------------------------------
__global__ __launch_bounds__(256) void pool_pow_kernel(
    const float* __restrict__ x,
    const float* __restrict__ l1p, const float* __restrict__ l2p,
    const float* __restrict__ gp,
    float* __restrict__ a_out, float* __restrict__ b_out)
{
    const int lane = threadIdx.x & 31;
    const int wave = threadIdx.x >> 5;
    const int w    = blockIdx.x * 8 + wave;       // global wave id
    const int b    = w / TILES_PER_B;
    const int tile = w - b * TILES_PER_B;
    const int n0   = tile * 16;
    const int ln   = lane & 15;
    const int half = lane >> 4;

    const float l1 = softplus_f(l1p[0]);
    const float l2 = softplus_f(l2p[0]);
    const float g  = softplus_f(gp[0]);

    const float* xb = x + (size_t)b * Cc * NT;
    const int col  = n0 + ln;
    const int colc = (col < NT) ? col : (NT - 1);   // clamp: masked cols are junk, never stored

    // W rows: M=0 averages channels 0..31, M=1 averages 32..63 (weight 1/32)
    const float aval_lo = (ln == 0) ? 0.03125f : 0.0f; // used while channels < 32
    const float aval_hi = (ln == 1) ? 0.03125f : 0.0f; // used while channels >= 32

    v8f acc = {0.f, 0.f, 0.f, 0.f, 0.f, 0.f, 0.f, 0.f};
    #pragma unroll
    for (int kk = 0; kk < 16; ++kk) {              // K-steps of 4 cover all 64 channels
        const int c0 = kk * 4 + 2 * half;          // channel for A/B element K = 0 + 2*half
        const float av = (kk < 8) ? aval_lo : aval_hi;
        v2f A; A.x = av; A.y = av;
        v2f Bv;
        Bv.x = xb[(size_t)c0 * NT + colc];
        Bv.y = xb[(size_t)(c0 + 1) * NT + colc];
        acc = __builtin_amdgcn_wmma_f32_16x16x4_f32(
            /*neg_a=*/false, A, /*neg_b=*/false, Bv,
            /*c_mod=*/(short)0, acc, /*reuse_a=*/false, /*reuse_b=*/false);
    }

    // Lanes 0-15 hold pooled row 0 in acc[0] and row 1 in acc[1].
    if (half == 0 && col < NT) {
        const float v1 = softplus_f(acc[0]) + 1.0f; // >= 1, safe for pow
        const float v2 = softplus_f(acc[1]) + 1.0f;
        const float av = g * expf(l1 * logf(v1));   // row factor, g folded in
        const float bv = expf(l2 * logf(v2));       // column factor
        const int n = col / Tt;
        const int t = col - n * Tt;                  // col = n*T + t (x memory order)
        const int idx = (b * Tt + t) * Nn + n;       // [bt, n] order
        a_out[idx] = av;
        b_out[idx] = bv;
    }
}

// ---------------------------------------------------------------------------
// Phase 1b: d2[i,j] = dist[i,j]^(2*l3)
// ---------------------------------------------------------------------------
__global__ __launch_bounds__(256) void dist2_kernel(
    const float* __restrict__ dist, const float* __restrict__ l3p,
    float* __restrict__ d2)
{
    const int i = blockIdx.x * 256 + threadIdx.x;
    if (i < Nn * Nn) {
        const float l3 = softplus_f(l3p[0]);
        d2[i] = expf(2.0f * l3 * logf(dist[i]));    // dist in [0.1, 2], positive
    }
}

// ---------------------------------------------------------------------------
// Phase 2: one wave32 per softmax row (bt, i); 325 values live in 11 VGPRs/lane.
// val = a_scaled[bt,i] * b2[bt,j] * d2[i,j]; single write of the output.
// ---------------------------------------------------------------------------
__global__ __launch_bounds__(256) void softmax_kernel(
    const float* __restrict__ a_s, const float* __restrict__ b2,
    const float* __restrict__ d2, float* __restrict__ out)
{
    const int lane = threadIdx.x & 31;
    const int wave = threadIdx.x >> 5;
    const int row  = blockIdx.x * 8 + wave;        // < 249600
    const int bt   = row / Nn;
    const int i    = row - bt * Nn;

    const float arow = a_s[row];
    const float* __restrict__ dr = d2 + (size_t)i * Nn;
    const float* __restrict__ br = b2 + (size_t)bt * Nn;

    // d2 row is shared by 768 bt's; keep it hot (global_prefetch_b8)
    __builtin_prefetch(dr + lane * 16, 0, 3);

    float vals[11];
    float m = -__builtin_inff();
    #pragma unroll
    for (int k = 0; k < 11; ++k) {
        const int j = lane + 32 * k;
        float v = -__builtin_inff();
        if (j < Nn) v = arow * br[j] * dr[j];
        vals[k] = v;
        m = fmaxf(m, v);
    }
    #pragma unroll
    for (int off = 16; off > 0; off >>= 1)
        m = fmaxf(m, __shfl_xor(m, off, 32));

    float s = 0.0f;
    #pragma unroll
    for (int k = 0; k < 11; ++k) {
        const int j = lane + 32 * k;
        if (j < Nn) {
            const float e = expf(vals[k] - m);
            vals[k] = e;
            s += e;
        }
    }
    #pragma unroll
    for (int off = 16; off > 0; off >>= 1)
        s += __shfl_xor(s, off, 32);
    const float inv = 1.0f / s;

    float* __restrict__ orow = out + (size_t)row * Nn;
    #pragma unroll
    for (int k = 0; k < 11; ++k) {
        const int j = lane + 32 * k;
        if (j < Nn) orow[j] = vals[k] * inv;
    }
}

// ---------------------------------------------------------------------------
extern "C" void kernel_launch(void* const* d_in, const int* in_sizes, int n_in,
                              void* d_out, int out_size, void* d_ws, size_t ws_size,
                              hipStream_t stream)
{
    const float* x    = (const float*)d_in[0]; // [64,64,325,12]
    const float* dist = (const float*)d_in[1]; // [325,325]
    const float* l1   = (const float*)d_in[2];
    const float* l2   = (const float*)d_in[3];
    const float* l3   = (const float*)d_in[4];
    const float* G    = (const float*)d_in[5];

    float* ws  = (float*)d_ws;
    float* a_s = ws;               // ROWS floats: g * x1^l1 in [bt, n] order
    float* b2  = ws + ROWS;        // ROWS floats: x2^l2 in [bt, n] order
    float* d2  = ws + 2 * ROWS;    // N*N floats: dist^(2*l3)
    float* out = (float*)d_out;    // [64,12,325,325]

    pool_pow_kernel<<<(Bn * TILES_PER_B) / 8, 256, 0, stream>>>(x, l1, l2, G, a_s, b2);
    dist2_kernel<<<(Nn * Nn + 255) / 256, 256, 0, stream>>>(dist, l3, d2);
    softmax_kernel<<<ROWS / 8, 256, 0, stream>>>(a_s, b2, d2, out);
}